// CoreProcessor_22849226014972
// MI455X (gfx1250) — compile-verified
//
#include <hip/hip_runtime.h>

// ---------------------------------------------------------------------------
// CDNA5 / gfx1250, wave32. All matrix work uses V_WMMA_F32_16X16X4_F32
// (full fp32 precision; 16x16 C tile = 8 VGPRs/lane, A/B frags = 2 VGPRs/lane).
// ---------------------------------------------------------------------------

typedef __attribute__((ext_vector_type(2))) float v2f;
typedef __attribute__((ext_vector_type(8))) float v8f;

#define NB 4
#define NS 1024
#define ND 128
#define NM 20000
#define NR (NB * NS)            // 4096 rows
#define NPART ((NM + 255) / 256) // 79 partial blocks per batch

__device__ __forceinline__ float wave_reduce_add(float v) {
#pragma unroll
    for (int m = 16; m >= 1; m >>= 1) v += __shfl_xor(v, m, 32);
    return v;
}

__device__ __forceinline__ v8f wmma_f32(v2f a, v2f b, v8f c) {
    return __builtin_amdgcn_wmma_f32_16x16x4_f32(
        /*neg_a=*/false, a, /*neg_b=*/false, b,
        /*c_mod=*/(short)0, c, /*reuse_a=*/false, /*reuse_b=*/false);
}

// --------------------------- row L2-normalize ------------------------------
__global__ void normalize_rows_kernel(const float* __restrict__ in,
                                      float* __restrict__ out,
                                      float* __restrict__ ns_out) {
    const int row  = blockIdx.x;
    const int lane = threadIdx.x;
    const float4 v = ((const float4*)(in + (size_t)row * ND))[lane];
    const float sq = v.x * v.x + v.y * v.y + v.z * v.z + v.w * v.w;
    const float ns = wave_reduce_add(sq);
    const float r  = rsqrtf(ns + 1e-12f);
    float4 o;
    o.x = v.x * r; o.y = v.y * r; o.z = v.z * r; o.w = v.w * r;
    ((float4*)(out + (size_t)row * ND))[lane] = o;
    if (ns_out != nullptr && lane == 0) ns_out[row] = ns;
}

// ------------------ LDS-tiled transpose: memT[d,m] = mem[m,d] --------------
__global__ void transpose_kernel(const float* __restrict__ in,
                                 float* __restrict__ outT) {
    __shared__ float tile[32][33];
    const int m0 = blockIdx.x * 32;
    const int d0 = blockIdx.y * 32;
    const int tx = threadIdx.x;   // 32
    const int ty = threadIdx.y;   // 8
#pragma unroll
    for (int i = 0; i < 32; i += 8)
        tile[ty + i][tx] = in[(size_t)(m0 + ty + i) * ND + d0 + tx];
    __syncthreads();
#pragma unroll
    for (int i = 0; i < 32; i += 8)
        outT[(size_t)(d0 + ty + i) * NM + m0 + tx] = tile[tx][ty + i];
}

// ------------------- generic  C = A * B^T + bias  (K=128) ------------------
__global__ void wmma_gemm_bias_kernel(const float* __restrict__ A,
                                      const float* __restrict__ Bw,
                                      const float* __restrict__ bias,
                                      float* __restrict__ C) {
    const int lane = threadIdx.x;
    const int l    = lane & 15;
    const int h2   = (lane >> 4) << 1;   // 0 or 2 (K sub-offset)
    const int row0 = blockIdx.x * 16;
    const int col0 = blockIdx.y * 16;
    const float* arow = A  + (size_t)(row0 + l) * ND;
    const float* brow = Bw + (size_t)(col0 + l) * ND;
    v8f acc = {};
#pragma unroll
    for (int k0 = 0; k0 < ND; k0 += 4) {
        v2f a = *(const v2f*)(arow + k0 + h2);
        v2f b = *(const v2f*)(brow + k0 + h2);
        acc = wmma_f32(a, b, acc);
    }
    const int   half = lane >> 4;
    const float bval = bias ? bias[col0 + l] : 0.0f;
#pragma unroll
    for (int r = 0; r < 8; ++r) {
        const int row = row0 + r + 8 * half;
        C[(size_t)row * ND + col0 + l] = acc[r] + bval;
    }
}

// ----------------------- LayerNorm + ReLU (in place) -----------------------
__global__ void ln_relu_kernel(float* __restrict__ h,
                               const float* __restrict__ gamma,
                               const float* __restrict__ beta) {
    const int row  = blockIdx.x;
    const int lane = threadIdx.x;
    float4 v = ((const float4*)(h + (size_t)row * ND))[lane];
    const float s   = wave_reduce_add(v.x + v.y + v.z + v.w);
    const float sq  = wave_reduce_add(v.x * v.x + v.y * v.y + v.z * v.z + v.w * v.w);
    const float mu  = s * (1.0f / (float)ND);
    const float var = sq * (1.0f / (float)ND) - mu * mu;
    const float rs  = rsqrtf(var + 1e-5f);
    const float4 gv = ((const float4*)gamma)[lane];
    const float4 bv = ((const float4*)beta)[lane];
    float4 o;
    o.x = fmaxf((v.x - mu) * rs * gv.x + bv.x, 0.0f);
    o.y = fmaxf((v.y - mu) * rs * gv.y + bv.y, 0.0f);
    o.z = fmaxf((v.z - mu) * rs * gv.z + bv.z, 0.0f);
    o.w = fmaxf((v.w - mu) * rs * gv.w + bv.w, 0.0f);
    ((float4*)(h + (size_t)row * ND))[lane] = o;
}

// ----- sim = xn * mn^T, thresholded + reduced over S, fused into cw[b,m] ---
// 1 wave per (b, 32-wide m tile): two mn tiles resident in VGPRs, every A
// fragment feeds two WMMAs (halves load traffic per matrix op).
__global__ void sim_cw_kernel(const float* __restrict__ xn,
                              const float* __restrict__ mn,
                              float* __restrict__ cw) {
    const int lane = threadIdx.x;
    const int l    = lane & 15;
    const int h2   = (lane >> 4) << 1;
    const int col0 = blockIdx.x * 32;    // first of two 16-wide m tiles
    const int b    = blockIdx.y;
    const float* A     = xn + (size_t)b * NS * ND;
    const float* brow0 = mn + (size_t)(col0 + l) * ND;
    const float* brow1 = mn + (size_t)(col0 + 16 + l) * ND;

    v2f bf0[32], bf1[32];
#pragma unroll
    for (int k = 0; k < 32; ++k) {
        bf0[k] = *(const v2f*)(brow0 + k * 4 + h2);
        bf1[k] = *(const v2f*)(brow1 + k * 4 + h2);
    }

    float csum0 = 0.0f, csum1 = 0.0f;
    for (int s0 = 0; s0 < NS; s0 += 16) {
        const float* arow = A + (size_t)(s0 + l) * ND;
        v8f acc0 = {}, acc1 = {};
#pragma unroll
        for (int k = 0; k < 32; ++k) {
            v2f a = *(const v2f*)(arow + k * 4 + h2);
            acc0 = wmma_f32(a, bf0[k], acc0);
            acc1 = wmma_f32(a, bf1[k], acc1);
        }
#pragma unroll
        for (int r = 0; r < 8; ++r) {
            const float s0v = acc0[r];
            const float s1v = acc1[r];
            csum0 += (s0v > 0.1f) ? s0v : 0.0f;   // THRESH gate
            csum1 += (s1v > 0.1f) ? s1v : 0.0f;
        }
    }
    csum0 += __shfl_xor(csum0, 16, 32);           // combine row-halves
    csum1 += __shfl_xor(csum1, 16, 32);
    if (lane < 16) {
        cw[(size_t)b * NM + col0 + l]      = csum0;
        cw[(size_t)b * NM + col0 + 16 + l] = csum1;
    }
}

// --------- gating: g[b,m] = active ? cw/ns : 0 ; per-block eff partials ----
__global__ void eff_kernel(const float* __restrict__ cw,
                           const float* __restrict__ ns,
                           float* __restrict__ g,
                           float* __restrict__ tpart) {
    const int b = blockIdx.y;
    const int m = blockIdx.x * 256 + threadIdx.x;
    float eff = 0.0f;
    if (m < NM) {
        const float c = cw[(size_t)b * NM + m];
        const float n = ns[m];
        const bool active = (c > 0.01f) && (n > 1e-6f);
        eff = active ? c : 0.0f;
        g[(size_t)b * NM + m] = active ? (c / n) : 0.0f;
    }
    __shared__ float red[8];
    const float w = wave_reduce_add(eff);
    if ((threadIdx.x & 31) == 0) red[threadIdx.x >> 5] = w;
    __syncthreads();
    if (threadIdx.x < 8) {
        float t = red[threadIdx.x];
#pragma unroll
        for (int mm = 4; mm >= 1; mm >>= 1) t += __shfl_xor(t, mm, 32);
        if (threadIdx.x == 0) tpart[(size_t)b * NPART + blockIdx.x] = t;
    }
}

// ------------- ti[b] = fixed-order sum of partials (deterministic) ---------
__global__ void ti_finalize_kernel(const float* __restrict__ tpart,
                                   float* __restrict__ ti) {
    const int b = blockIdx.x;
    float t = 0.0f;
    for (int i = threadIdx.x; i < NPART; i += 32)
        t += tpart[(size_t)b * NPART + i];
    t = wave_reduce_add(t);
    if (threadIdx.x == 0) ti[b] = t;
}

// -------- G_b[d1,d2] = sum_m g[b,m] * mem[m,d1] * mem[m,d2]  (WMMA) --------
// memT layout makes all fragment loads contiguous b64.
__global__ void gram_kernel(const float* __restrict__ memT,
                            const float* __restrict__ g,
                            float* __restrict__ G) {
    const int lane = threadIdx.x;
    const int l    = lane & 15;
    const int h2   = (lane >> 4) << 1;
    const int d1   = blockIdx.x * 16;
    const int d2   = blockIdx.y * 16;
    const int b    = blockIdx.z;
    const float* gb = g + (size_t)b * NM;
    const float* a0 = memT + (size_t)(d1 + l) * NM;
    const float* b0 = memT + (size_t)(d2 + l) * NM;
    v8f acc = {};
#pragma unroll 8
    for (int m0 = 0; m0 < NM; m0 += 4) {
        v2f gg = *(const v2f*)(gb + m0 + h2);
        v2f av = *(const v2f*)(a0 + m0 + h2);
        v2f bv = *(const v2f*)(b0 + m0 + h2);
        av.x *= gg.x;
        av.y *= gg.y;
        acc = wmma_f32(av, bv, acc);
    }
    const int half = lane >> 4;
    float* Gb = G + (size_t)b * ND * ND;
#pragma unroll
    for (int r = 0; r < 8; ++r)
        Gb[(size_t)(d1 + r + 8 * half) * ND + d2 + l] = acc[r];
}

// -------- out = raw + scale_b * (raw @ G_b),  scale_b gated on ti ----------
__global__ void apply_kernel(const float* __restrict__ raw,
                             const float* __restrict__ G,
                             const float* __restrict__ ti,
                             float* __restrict__ out) {
    const int lane = threadIdx.x;
    const int l    = lane & 15;
    const int h2   = (lane >> 4) << 1;
    const int row0 = blockIdx.z * NS + blockIdx.x * 16;
    const int col0 = blockIdx.y * 16;
    const int b    = blockIdx.z;
    const float* arow = raw + (size_t)(row0 + l) * ND;
    const float* brow = G + (size_t)b * ND * ND + (size_t)(col0 + l) * ND; // G symmetric
    v8f acc = {};
#pragma unroll
    for (int k0 = 0; k0 < ND; k0 += 4) {
        v2f a  = *(const v2f*)(arow + k0 + h2);
        v2f bb = *(const v2f*)(brow + k0 + h2);
        acc = wmma_f32(a, bb, acc);
    }
    const float t     = ti[b];
    const float scale = (t > 0.01f) ? (0.5f / (t + 1e-5f)) : 0.0f;
    const int   half  = lane >> 4;
#pragma unroll
    for (int r = 0; r < 8; ++r) {
        const size_t idx = (size_t)(row0 + r + 8 * half) * ND + col0 + l;
        out[idx] = raw[idx] + scale * acc[r];
    }
}

// ---------------------------------------------------------------------------
extern "C" void kernel_launch(void* const* d_in, const int* in_sizes, int n_in,
                              void* d_out, int out_size, void* d_ws, size_t ws_size,
                              hipStream_t stream) {
    const float* x    = (const float*)d_in[0];  // [B,S,D]
    const float* mem  = (const float*)d_in[1];  // [M,D]
    const float* W1   = (const float*)d_in[2];  // [D,D]
    const float* b1   = (const float*)d_in[3];
    const float* ln_g = (const float*)d_in[4];
    const float* ln_b = (const float*)d_in[5];
    const float* W2   = (const float*)d_in[6];
    const float* b2   = (const float*)d_in[7];
    float* out = (float*)d_out;

    float* ws    = (float*)d_ws;
    float* xn    = ws;                      // NR*ND      = 524288
    float* mn    = xn + (size_t)NR * ND;    // NM*ND      = 2560000
    float* memT  = mn + (size_t)NM * ND;    // ND*NM      = 2560000
    float* ns    = memT + (size_t)ND * NM;  // NM         = 20000
    float* h     = ns + NM;                 // NR*ND (feats in-place)
    float* raw   = h + (size_t)NR * ND;     // NR*ND
    float* cw    = raw + (size_t)NR * ND;   // NB*NM
    float* g     = cw + (size_t)NB * NM;    // NB*NM
    float* tpart = g + (size_t)NB * NM;     // NB*NPART (316) -> pad 320
    float* ti    = tpart + 320;             // NB
    float* G     = ti + NB;                 // NB*ND*ND   = 65536

    // normalize query rows and memory rows (mem also yields squared norms)
    normalize_rows_kernel<<<NR, 32, 0, stream>>>(x, xn, nullptr);
    normalize_rows_kernel<<<NM, 32, 0, stream>>>(mem, mn, ns);

    // memT = mem^T for coalesced Gram fragments
    transpose_kernel<<<dim3(NM / 32, ND / 32), dim3(32, 8), 0, stream>>>(mem, memT);

    // fusion net: h = x@W1^T + b1 ; LN+ReLU ; raw = feats@W2^T + b2
    wmma_gemm_bias_kernel<<<dim3(NR / 16, ND / 16), 32, 0, stream>>>(x, W1, b1, h);
    ln_relu_kernel<<<NR, 32, 0, stream>>>(h, ln_g, ln_b);
    wmma_gemm_bias_kernel<<<dim3(NR / 16, ND / 16), 32, 0, stream>>>(h, W2, b2, raw);

    // cosine-sim GEMM fused with threshold + seq reduction -> cw[b,m]
    sim_cw_kernel<<<dim3(NM / 32, NB), 32, 0, stream>>>(xn, mn, cw);

    // gating weights + deterministic total influence
    eff_kernel<<<dim3(NPART, NB), 256, 0, stream>>>(cw, ns, g, tpart);
    ti_finalize_kernel<<<NB, 32, 0, stream>>>(tpart, ti);

    // per-batch Gram matrix G_b = mem^T diag(g_b) mem
    gram_kernel<<<dim3(ND / 16, ND / 16, NB), 32, 0, stream>>>(memT, g, G);

    // final = raw + 0.5 * (raw@G_b)/(ti+eps), gated on ti
    apply_kernel<<<dim3(NS / 16, ND / 16, NB), 32, 0, stream>>>(raw, G, ti, out);
}